// TemporalEncoder_40733469835951
// MI455X (gfx1250) — compile-verified
//
#include <hip/hip_runtime.h>

// CDNA5 / gfx1250, wave32. Fused gather + fp32 WMMA GEMM + LSTM-cell
// activations for the degenerate L=1, h0=c0=0 case of the reference.

typedef __attribute__((ext_vector_type(2))) float v2f;
typedef __attribute__((ext_vector_type(8))) float v8f;

#define B_SZ 2048
#define T_SZ 200
#define F_SZ 64
#define H_SZ 128

__device__ __forceinline__ float sigmoid_f(float x) {
    return 1.0f / (1.0f + __expf(-x));
}

__global__ void __launch_bounds__(32)
temporal_encoder_wmma(const float* __restrict__ x,      // [B, T, F]
                      const int*   __restrict__ x_len,  // [B]
                      const float* __restrict__ W_ih,   // [4H, F] row-major
                      const float* __restrict__ b_ih,   // [4H]
                      const float* __restrict__ b_hh,   // [4H]
                      float* __restrict__ h_out,        // [B, H]
                      float* __restrict__ c_out)        // [B, H]
{
    const int lane   = threadIdx.x;     // 0..31, one wave per block
    const int laneLo = lane & 15;
    const int hiSel  = lane >> 4;       // 0 or 1
    const int Mbase  = blockIdx.x * 16; // batch tile
    const int jbase  = blockIdx.y * 16; // hidden-unit tile

    // ---- Gather A tile: A[m, k] = x[Mbase+m, idx(Mbase+m), k], 16x64 ----
    // A-matrix 16x4 f32 layout: lanes 0-15 hold K={0,1}, lanes 16-31 K={2,3}.
    const int b_a = Mbase + laneLo;
    const int xl  = x_len[b_a];
    const int idx = (xl == 0) ? 0 : (xl - 1);
    const float* xrow = x + ((size_t)b_a * T_SZ + idx) * F_SZ + hiSel * 2;

    v2f a[16];
#pragma unroll
    for (int kk = 0; kk < 16; ++kk)
        a[kk] = *(const v2f*)(xrow + kk * 4);

    // ---- One 16(batch) x 16(gate) tile of gates = A * W_ih^T + bias ----
    auto gate_tile = [&](int nbase) -> v8f {
        // B 4x16 f32: B[k][n] = W_ih[nbase+n][k]; lanes 0-15 -> K={0,1},
        // lanes 16-31 -> K={2,3}, N = lane&15. Consecutive k -> float2 load.
        const float* wrow = W_ih + (size_t)(nbase + laneLo) * F_SZ + hiSel * 2;
        v8f acc = {};
#pragma unroll
        for (int kk = 0; kk < 16; ++kk) {
            v2f bv = *(const v2f*)(wrow + kk * 4);
            // 8 args: (neg_a, A, neg_b, B, c_mod, C, reuse_a, reuse_b)
            acc = __builtin_amdgcn_wmma_f32_16x16x4_f32(
                false, a[kk], false, bv, (short)0, acc, false, false);
        }
        // C/D layout: every acc register has N = lane&15 -> scalar bias add.
        const float bias = b_ih[nbase + laneLo] + b_hh[nbase + laneLo];
#pragma unroll
        for (int r = 0; r < 8; ++r) acc[r] += bias;
        return acc;
    };

    // PyTorch gate order i,f,g,o at offsets 0,128,256,384. The f gate is
    // multiplied by c_prev == 0 exactly, so it is skipped (exact algebra).
    v8f gi = gate_tile(jbase);            // input gate
    v8f gg = gate_tile(2 * H_SZ + jbase); // cell gate
    v8f go = gate_tile(3 * H_SZ + jbase); // output gate

    // ---- Activations + scatter. acc[r] @ lane = (b = Mbase+r+8*hiSel,
    //      j = jbase + (lane&15)). h first, then c (d_out concat order). ----
    const int j = jbase + laneLo;
#pragma unroll
    for (int r = 0; r < 8; ++r) {
        const int b = Mbase + r + hiSel * 8;
        const float cv = sigmoid_f(gi[r]) * tanhf(gg[r]);
        const float hv = sigmoid_f(go[r]) * tanhf(cv);
        h_out[(size_t)b * H_SZ + j] = hv;
        c_out[(size_t)b * H_SZ + j] = cv;
    }
}

extern "C" void kernel_launch(void* const* d_in, const int* in_sizes, int n_in,
                              void* d_out, int out_size, void* d_ws, size_t ws_size,
                              hipStream_t stream) {
    (void)in_sizes; (void)n_in; (void)out_size; (void)d_ws; (void)ws_size;
    const float* x     = (const float*)d_in[0]; // encode_features [B,T,F]
    const int*   x_len = (const int*)  d_in[1]; // [B]
    const float* W_ih  = (const float*)d_in[2]; // [4H,F]
    // d_in[3] = W_hh: unused (h_prev == 0 exactly for L=1)
    const float* b_ih  = (const float*)d_in[4]; // [4H]
    const float* b_hh  = (const float*)d_in[5]; // [4H]

    float* h_out = (float*)d_out;                        // hs: [1,B,H]
    float* c_out = (float*)d_out + (size_t)B_SZ * H_SZ;  // cs: [1,B,H]

    dim3 grid(B_SZ / 16, H_SZ / 16); // 128 x 8 one-wave workgroups
    temporal_encoder_wmma<<<grid, dim3(32), 0, stream>>>(
        x, x_len, W_ih, b_ih, b_hh, h_out, c_out);
}